// GraphUnet_60954175865405
// MI455X (gfx1250) — compile-verified
//
#include <hip/hip_runtime.h>
#include <hip/hip_bf16.h>

typedef __attribute__((ext_vector_type(2))) float v2f;
typedef __attribute__((ext_vector_type(8))) float v8f;

#define GB 32        // batch
#define GN 1024      // nodes
#define GC 128       // C_IN
#define GF 64        // F0=F1=F2
#define GOUT 10
#define GEPS 1e-5f

// ---------------------------------------------------------------------------
// Degree / normalization: d[b,n] = rsqrt( mask[n]*sum_m(A[n,m]*mask[m]) + 1 + eps )
// One wave32 per row, 8 rows per 256-thread block, float4 (b128) streaming.
// ---------------------------------------------------------------------------
__global__ __launch_bounds__(256) void deg_kernel(const float* __restrict__ A,
                                                  const float* __restrict__ mask,
                                                  float* __restrict__ d) {
    int lane = threadIdx.x & 31;
    int wave = threadIdx.x >> 5;
    int row = blockIdx.x * 8 + wave;          // row in [0, B*N)
    int b = row >> 10;                        // /GN
    int n = row & (GN - 1);
    const float4* Arow = (const float4*)(A + ((size_t)b * GN + n) * GN);
    const float4* mb4  = (const float4*)(mask + (size_t)b * GN);
    float s = 0.f;
    for (int m = lane; m < GN / 4; m += 32) {
        float4 a = Arow[m];
        float4 w = mb4[m];
        s += a.x * w.x + a.y * w.y + a.z * w.z + a.w * w.w;
    }
#pragma unroll
    for (int off = 16; off > 0; off >>= 1) s += __shfl_down(s, off, 32);
    if (lane == 0) {
        float t = mask[(size_t)b * GN + n] * s + 1.0f;
        d[row] = rsqrtf(t + GEPS);
    }
}

// ---------------------------------------------------------------------------
// Fused gconv: H = relu( (D m A m D + D I D) @ X @ W + bias )
// CIN in {128,64}, ROWS = 2048/CIN. 256 threads = 8 waves.
// Phase 1: T = L@X via WMMA f32 16x16x4. Raw A chunks are double-buffered in
//          LDS via GLOBAL_LOAD_ASYNC_TO_LDS_B128 (DMA for chunk i+1 overlaps
//          scale+WMMA of chunk i); normalization applied once per block by a
//          cooperative scale pass (no per-wave redundant VALU in the hot loop).
// Phase 2: T -> LDS.   Phase 3: H = relu(T@W + b) via WMMA.
// ---------------------------------------------------------------------------
template <int CIN, int ROWS>
__global__ __launch_bounds__(256) void gconv_kernel(const float* __restrict__ A,
                                                    const float* __restrict__ X,
                                                    const float* __restrict__ W,
                                                    const float* __restrict__ bias,
                                                    const float* __restrict__ d,
                                                    const float* __restrict__ mask,
                                                    float* __restrict__ H) {
    __shared__ __align__(16) float Araw[2][ROWS][64];  // raw DMA targets (16B-aligned rows)
    __shared__ float Lsh[ROWS][68];   // scaled tile; +4 pad -> conflict-free column reads
    __shared__ float Tsh[ROWS][CIN + 1];
    __shared__ float cmb[2][64];      // d[m]*mask[m], double-buffered (computed 1 chunk ahead)
    __shared__ float rn[ROWS];        // d[n]*mask[n]
    __shared__ float dsq[ROWS];       // d[n]*d[n] (diagonal of L)

    const int tid  = threadIdx.x;
    const int lane = tid & 31;
    const int wave = tid >> 5;
    const int half = lane >> 4;       // 0: lanes 0-15, 1: lanes 16-31
    const int mrow = lane & 15;       // A-frag M index / B-frag N index
    const int n0 = blockIdx.x * ROWS;
    const int b  = blockIdx.y;

    constexpr int CW = CIN / 16;      // col-waves per row-subtile
    const int sub   = wave / CW;      // row-subtile (0..ROWS/16-1)
    const int cw    = wave % CW;
    const int rbase = sub * 16;
    const int c0    = cw * 16 + mrow; // X column this lane owns in B-frag

    constexpr int NCHUNK = (ROWS * 64) / (256 * 4);    // async B128 issues per thread

    // ---- prologue: row factors, cm for chunk 0, DMA chunk 0 into buffer 0 ----
    if (tid < ROWS) {
        float dn = d[(size_t)b * GN + n0 + tid];
        float mn = mask[(size_t)b * GN + n0 + tid];
        rn[tid]  = dn * mn;
        dsq[tid] = dn * dn;
    }
    if (tid < 64) {
        cmb[0][tid] = d[(size_t)b * GN + tid] * mask[(size_t)b * GN + tid];
    }
#pragma unroll
    for (int it = 0; it < NCHUNK; ++it) {
        int idx4 = (it * 256 + tid) * 4;
        int r = idx4 >> 6, k = idx4 & 63;
        const float* gsrc = A + ((size_t)b * GN + (n0 + r)) * GN + k;
        unsigned ldst = (unsigned)(size_t)&Araw[0][r][k];
        asm volatile("global_load_async_to_lds_b128 %0, %1, off"
                     :: "v"(ldst), "v"(gsrc) : "memory");
    }

    v8f acc = {};
    int buf = 0;

    for (int ci = 0; ci < GN / 64; ++ci) {
        const int m0 = ci * 64;
        asm volatile("s_wait_asynccnt 0x0" ::: "memory");  // Araw[buf] landed (this wave)
        __syncthreads();  // all DMA done; prior Lsh readers & cm/rn writers retired

        // prefetch chunk ci+1: cm + raw A DMA into the other buffer
        if (ci + 1 < GN / 64) {
            const int mn0 = m0 + 64;
            if (tid < 64) {
                cmb[buf ^ 1][tid] =
                    d[(size_t)b * GN + mn0 + tid] * mask[(size_t)b * GN + mn0 + tid];
            }
#pragma unroll
            for (int it = 0; it < NCHUNK; ++it) {
                int idx4 = (it * 256 + tid) * 4;
                int r = idx4 >> 6, k = idx4 & 63;
                const float* gsrc = A + ((size_t)b * GN + (n0 + r)) * GN + (mn0 + k);
                unsigned ldst = (unsigned)(size_t)&Araw[buf ^ 1][r][k];
                asm volatile("global_load_async_to_lds_b128 %0, %1, off"
                             :: "v"(ldst), "v"(gsrc) : "memory");
            }
        }

        // scale once per block: Lsh = rn*cm*Araw (+ diag), amortized over 256 threads
#pragma unroll
        for (int i = tid; i < ROWS * 64; i += 256) {
            int r = i >> 6, k = i & 63;
            float v = rn[r] * cmb[buf][k] * Araw[buf][r][k];
            if (n0 + r == m0 + k) v += dsq[r];
            Lsh[r][k] = v;
        }
        __syncthreads();  // Lsh ready block-wide

        const float* Xb = X + ((size_t)b * GN + m0) * CIN;
#pragma unroll
        for (int kk = 0; kk < 64; kk += 4) {
            int ka = kk + 2 * half;   // lanes 0-15: K0/K1, 16-31: K2/K3
            v2f af, bf;
            af.x = Lsh[rbase + mrow][ka];
            af.y = Lsh[rbase + mrow][ka + 1];
            const float* xp = Xb + (size_t)ka * CIN + c0;
            bf.x = xp[0];
            bf.y = xp[CIN];
            acc = __builtin_amdgcn_wmma_f32_16x16x4_f32(
                false, af, false, bf, (short)0, acc, false, false);
        }
        buf ^= 1;
    }

    __syncthreads();
    // Phase 2: spill T tile to LDS (C/D layout: VGPR v -> M = v + 8*half)
#pragma unroll
    for (int v = 0; v < 8; ++v)
        Tsh[rbase + v + 8 * half][cw * 16 + mrow] = acc[v];
    __syncthreads();

    // Phase 3: out = relu(T @ W + bias), tile ROWS x 64
    constexpr int ACT = (ROWS / 16) * 4;      // active waves
    if (wave < ACT) {
        const int sub2 = wave / 4;
        const int cw2  = wave % 4;
        const int rb2  = sub2 * 16;
        const int c2   = cw2 * 16 + mrow;
        v8f acc2 = {};
#pragma unroll
        for (int kk = 0; kk < CIN; kk += 4) {
            int ka = kk + 2 * half;
            v2f af, bf;
            af.x = Tsh[rb2 + mrow][ka];
            af.y = Tsh[rb2 + mrow][ka + 1];
            bf.x = W[(size_t)ka * GF + c2];
            bf.y = W[(size_t)(ka + 1) * GF + c2];
            acc2 = __builtin_amdgcn_wmma_f32_16x16x4_f32(
                false, af, false, bf, (short)0, acc2, false, false);
        }
        float bv = bias[c2];
#pragma unroll
        for (int v = 0; v < 8; ++v) {
            int n = n0 + rb2 + v + 8 * half;
            float h = acc2[v] + bv;
            H[((size_t)b * GN + n) * GF + c2] = fmaxf(h, 0.f);
        }
    }
}

// ---------------------------------------------------------------------------
// Top-k pool: score, stable ascending rank by counting, drop lowest scores,
// rescale x in place by tanh(y)*new_mask. One block (1024 thr) per graph.
// ---------------------------------------------------------------------------
__global__ __launch_bounds__(1024) void pool_kernel(float* __restrict__ x,
                                                    const float* __restrict__ mask_in,
                                                    const int* __restrict__ nn_in,
                                                    const float* __restrict__ p,
                                                    float* __restrict__ mask_out,
                                                    int* __restrict__ nn_out,
                                                    float rem_frac) {
    __shared__ float p_sh[GF];
    __shared__ float yeff[GN];
    const int b = blockIdx.x;
    const int t = threadIdx.x;
    if (t < GF) p_sh[t] = p[t];
    __syncthreads();

    float pn2 = 0.f;
#pragma unroll
    for (int c = 0; c < GF; ++c) pn2 += p_sh[c] * p_sh[c];
    float inv = rsqrtf(pn2);

    float* xr = x + ((size_t)b * GN + t) * GF;
    float acc = 0.f;
#pragma unroll
    for (int c = 0; c < GF; ++c) acc += xr[c] * p_sh[c];
    float y = acc * inv;
    float m = mask_in[(size_t)b * GN + t];
    float ye = (m > 0.f) ? y : __builtin_inff();
    yeff[t] = ye;
    __syncthreads();

    int rank = 0;
    for (int j = 0; j < GN; ++j) {
        float yj = yeff[j];
        rank += (int)((yj < ye) || (yj == ye && j < t));
    }
    int nnb = nn_in[b];
    int nrem = (int)((float)nnb * rem_frac);
    float nm = (rank >= nrem) ? m : 0.f;
    mask_out[(size_t)b * GN + t] = nm;
    if (t == 0) nn_out[b] = nnb - nrem;

    float scale = tanhf(y) * nm;
#pragma unroll
    for (int c = 0; c < GF; ++c) xr[c] = xr[c] * scale;
}

// ---------------------------------------------------------------------------
// Global max pool over nodes + final FC (64 -> 10). One block per graph.
// ---------------------------------------------------------------------------
__global__ __launch_bounds__(64) void finalize_kernel(const float* __restrict__ x,
                                                      const float* __restrict__ Wfc,
                                                      const float* __restrict__ bfc,
                                                      float* __restrict__ out) {
    __shared__ float mx[GF];
    const int b = blockIdx.x;
    const int c = threadIdx.x;
    float mv = -__builtin_inff();
    const float* xb = x + (size_t)b * GN * GF + c;
    for (int n = 0; n < GN; ++n) mv = fmaxf(mv, xb[(size_t)n * GF]);
    mx[c] = mv;
    __syncthreads();
    if (c < GOUT) {
        float s = bfc[c];
#pragma unroll
        for (int k = 0; k < GF; ++k) s += mx[k] * Wfc[(size_t)k * GOUT + c];
        out[(size_t)b * GOUT + c] = s;
    }
}

// ---------------------------------------------------------------------------
extern "C" void kernel_launch(void* const* d_in, const int* in_sizes, int n_in,
                              void* d_out, int out_size, void* d_ws, size_t ws_size,
                              hipStream_t stream) {
    const float* x    = (const float*)d_in[0];
    const float* A    = (const float*)d_in[1];
    const float* mask = (const float*)d_in[2];
    const int*   nn0  = (const int*)d_in[3];
    const float* W0   = (const float*)d_in[4];
    const float* b0   = (const float*)d_in[5];
    const float* W1   = (const float*)d_in[6];
    const float* b1   = (const float*)d_in[7];
    const float* W2   = (const float*)d_in[8];
    const float* b2   = (const float*)d_in[9];
    const float* p0   = (const float*)d_in[10];
    const float* p1   = (const float*)d_in[11];
    const float* Wfc  = (const float*)d_in[12];
    const float* bfc  = (const float*)d_in[13];
    float* out = (float*)d_out;

    // workspace carve-up
    float* ws = (float*)d_ws;
    float* dvec = ws;                 ws += GB * GN;   // 32768
    float* m1   = ws;                 ws += GB * GN;
    float* m2   = ws;                 ws += GB * GN;
    int*   nn1  = (int*)ws;           ws += 32;
    int*   nn2  = (int*)ws;           ws += 32;
    float* h1   = ws;                 ws += (size_t)GB * GN * GF;  // 8 MB
    float* h2   = ws;                 ws += (size_t)GB * GN * GF;
    float* h3   = ws;

    const float REMF = 1.0f - 0.8f;   // fraction removed per pool

    dim3 blk256(256), blk1024(1024), blk64(64);
    dim3 gDeg(GB * GN / 8);
    dim3 gConv0(GN / 16, GB);         // CIN=128, ROWS=16
    dim3 gConvF(GN / 32, GB);         // CIN=64,  ROWS=32
    dim3 gB(GB);

    // conv0 (A L2-resident after first pass; masks applied on the fly)
    deg_kernel<<<gDeg, blk256, 0, stream>>>(A, mask, dvec);
    gconv_kernel<GC, 16><<<gConv0, blk256, 0, stream>>>(A, x, W0, b0, dvec, mask, h1);
    pool_kernel<<<gB, blk1024, 0, stream>>>(h1, mask, nn0, p0, m1, nn1, REMF);

    // conv1 (cumulative masks: m1*A*m1 == current masked adjacency)
    deg_kernel<<<gDeg, blk256, 0, stream>>>(A, m1, dvec);
    gconv_kernel<GF, 32><<<gConvF, blk256, 0, stream>>>(A, h1, W1, b1, dvec, m1, h2);
    pool_kernel<<<gB, blk1024, 0, stream>>>(h2, m1, nn1, p1, m2, nn2, REMF);

    // conv2
    deg_kernel<<<gDeg, blk256, 0, stream>>>(A, m2, dvec);
    gconv_kernel<GF, 32><<<gConvF, blk256, 0, stream>>>(A, h2, W2, b2, dvec, m2, h3);

    finalize_kernel<<<gB, blk64, 0, stream>>>(h3, Wfc, bfc, out);
}